// DGCNN_13314398618268
// MI455X (gfx1250) — compile-verified
//
#include <hip/hip_runtime.h>
#include <math.h>

#define F_INPUT 128
#define CH 32
#define NPG_C 100
#define KTOP 30
#define NCAT 97
#define NCLS 10

typedef __attribute__((ext_vector_type(2))) float v2f;
typedef __attribute__((ext_vector_type(8))) float v8f;

// ---------------- utility ----------------
__global__ void k_zero(float* __restrict__ p, int n) {
    int i = blockIdx.x * 256 + threadIdx.x;
    if (i < n) p[i] = 0.0f;
}

__global__ void k_deg(const int* __restrict__ dst, float* __restrict__ deg, int E) {
    int e = blockIdx.x * 256 + threadIdx.x;
    if (e < E) atomicAdd(&deg[dst[e]], 1.0f);
}

__global__ void k_dinv(float* __restrict__ d, int n) {
    int i = blockIdx.x * 256 + threadIdx.x;
    if (i < n) d[i] = rsqrtf(d[i] + 1.0f);   // deg + self-loop
}

// ---------------- WMMA GEMM: C[M x 32] = A[M x K] * W[K x 32] ----------------
// One wave computes a 16-row x 32-col tile with two f32 16x16x4 WMMA accumulators.
// A frag (16x4 f32): lanes 0-15 hold K=k0,k0+1 ; lanes 16-31 hold K=k0+2,k0+3 (M = lane&15).
// B frag (4x16 f32): VGPR0 = rows k0 (lanes 0-15) / k0+2 (lanes 16-31), VGPR1 = k0+1 / k0+3.
// D (16x16 f32): VGPR v -> row v + 8*(lane>>4), col = lane&15.
__global__ __launch_bounds__(256) void k_gemm_n32(const float* __restrict__ A,
                                                  const float* __restrict__ W,
                                                  float* __restrict__ C,
                                                  int M, int K) {
    __shared__ float wlds[F_INPUT * CH];
    const int tid = threadIdx.x;
    for (int i = tid; i < K * CH; i += 256) wlds[i] = W[i];
    __syncthreads();

    const int wave = tid >> 5;
    const int lane = tid & 31;
    const int row0 = blockIdx.x * 128 + wave * 16;
    const int m    = lane & 15;
    const int kh   = lane >> 4;          // 0 or 1

    v8f acc0 = {0.f, 0.f, 0.f, 0.f, 0.f, 0.f, 0.f, 0.f};
    v8f acc1 = {0.f, 0.f, 0.f, 0.f, 0.f, 0.f, 0.f, 0.f};

    int r = row0 + m;
    if (r > M - 1) r = M - 1;            // clamp (no EXEC divergence before WMMA)
    const float* __restrict__ arow = A + (size_t)r * K;

    for (int k0 = 0; k0 < K; k0 += 4) {
        v2f a;
        a.x = arow[k0 + 2 * kh];
        a.y = arow[k0 + 2 * kh + 1];
        const float* bp = wlds + (k0 + 2 * kh) * CH;
        v2f b0, b1;
        b0.x = bp[m];       b0.y = bp[CH + m];
        b1.x = bp[16 + m];  b1.y = bp[CH + 16 + m];
        acc0 = __builtin_amdgcn_wmma_f32_16x16x4_f32(false, a, false, b0, (short)0, acc0, false, false);
        acc1 = __builtin_amdgcn_wmma_f32_16x16x4_f32(false, a, false, b1, (short)0, acc1, false, false);
    }

#pragma unroll
    for (int v = 0; v < 8; v++) {
        int rr = row0 + v + 8 * kh;
        if (rr < M) {
            C[(size_t)rr * CH + m]      = acc0[v];
            C[(size_t)rr * CH + 16 + m] = acc1[v];
        }
    }
}

// ---------------- GCN aggregation (32 channels) ----------------
__global__ void k_init32(const float* __restrict__ H, const float* __restrict__ dinv,
                         float* __restrict__ agg, int N) {
    int idx = blockIdx.x * 256 + threadIdx.x;
    if (idx >= N * CH) return;
    int i = idx >> 5;
    float di = dinv[i];
    agg[idx] = H[idx] * di * di;         // self-loop term
}

__global__ void k_scatter32(const int* __restrict__ src, const int* __restrict__ dst,
                            const float* __restrict__ dinv, const float* __restrict__ H,
                            float* __restrict__ agg, int E) {
    int e = blockIdx.x * 256 + threadIdx.x;
    if (e >= E) return;
    int s = src[e], d = dst[e];
    float nrm = dinv[s] * dinv[d];
    const float4* __restrict__ hp = (const float4*)(H + (size_t)s * CH);
    float* ap = agg + (size_t)d * CH;
#pragma unroll
    for (int q = 0; q < CH / 4; q++) {
        float4 h = hp[q];
        atomicAdd(ap + 4 * q + 0, h.x * nrm);
        atomicAdd(ap + 4 * q + 1, h.y * nrm);
        atomicAdd(ap + 4 * q + 2, h.z * nrm);
        atomicAdd(ap + 4 * q + 3, h.w * nrm);
    }
}

__global__ void k_fin32(const float* __restrict__ agg, const float* __restrict__ b,
                        float* __restrict__ xd, float* __restrict__ xc, int N, int xcoff) {
    int idx = blockIdx.x * 256 + threadIdx.x;
    if (idx >= N * CH) return;
    int i = idx >> 5, c = idx & 31;
    float t = tanhf(agg[idx] + b[c]);
    xd[idx] = t;
    xc[(size_t)i * NCAT + xcoff + c] = t;
}

// ---------------- layer 4 (1 channel) ----------------
__global__ void k_h4(const float* __restrict__ xd, const float* __restrict__ W4,
                     float* __restrict__ h4, int N) {
    int i = blockIdx.x * 256 + threadIdx.x;
    if (i >= N) return;
    const float* r = xd + (size_t)i * CH;
    float s = 0.f;
#pragma unroll
    for (int c = 0; c < CH; c++) s += r[c] * W4[c];
    h4[i] = s;
}

__global__ void k_init4(const float* __restrict__ h4, const float* __restrict__ dinv,
                        float* __restrict__ agg4, int N) {
    int i = blockIdx.x * 256 + threadIdx.x;
    if (i >= N) return;
    float di = dinv[i];
    agg4[i] = h4[i] * di * di;
}

__global__ void k_scatter4(const int* __restrict__ src, const int* __restrict__ dst,
                           const float* __restrict__ dinv, const float* __restrict__ h4,
                           float* __restrict__ agg4, int E) {
    int e = blockIdx.x * 256 + threadIdx.x;
    if (e >= E) return;
    int s = src[e], d = dst[e];
    atomicAdd(&agg4[d], h4[s] * dinv[s] * dinv[d]);
}

__global__ void k_fin4(const float* __restrict__ agg4, const float* __restrict__ b4,
                       float* __restrict__ xc, int N) {
    int i = blockIdx.x * 256 + threadIdx.x;
    if (i >= N) return;
    xc[(size_t)i * NCAT + 96] = tanhf(agg4[i] + b4[0]);
}

// ---------------- global_sort_pool ----------------
// Rank = #{v_j > v_i} + #{v_j == v_i, j < i}  (matches stable argsort of -v).
__global__ __launch_bounds__(128) void k_sortpool(const float* __restrict__ xc,
                                                  float* __restrict__ topk) {
    __shared__ float vals[NPG_C];
    int g = blockIdx.x;
    int tid = threadIdx.x;
    for (int i = tid; i < NPG_C; i += 128)
        vals[i] = xc[(size_t)(g * NPG_C + i) * NCAT + 96];
    __syncthreads();
    for (int i = tid; i < NPG_C; i += 128) {
        float v = vals[i];
        int rank = 0;
        for (int j = 0; j < NPG_C; j++) {
            float u = vals[j];
            rank += (u > v) || (u == v && j < i);
        }
        if (rank < KTOP) {
            const float* sp = xc + (size_t)(g * NPG_C + i) * NCAT;
            float* dp = topk + ((size_t)g * KTOP + rank) * NCAT;
            for (int c = 0; c < NCAT; c++) dp[c] = sp[c];
        }
    }
}

// ---------------- conv1d(1,16,97,stride 97) + relu -> [G,16,30] ----------------
__global__ void k_conv1(const float* __restrict__ topk, const float* __restrict__ w,
                        const float* __restrict__ b, float* __restrict__ out, int G) {
    int idx = blockIdx.x * 256 + threadIdx.x;
    if (idx >= G * 16 * KTOP) return;
    int g = idx / (16 * KTOP);
    int rem = idx - g * 16 * KTOP;
    int f = rem / KTOP, t = rem - f * KTOP;
    const float* zp = topk + ((size_t)g * KTOP + t) * NCAT;
    const float* wp = w + f * NCAT;
    float s = b[f];
    for (int i = 0; i < NCAT; i++) s += zp[i] * wp[i];
    out[(size_t)g * 16 * KTOP + f * KTOP + t] = fmaxf(s, 0.f);
}

// ---------------- maxpool(2,2) -> [G,16,15] ----------------
__global__ void k_pool(const float* __restrict__ in, float* __restrict__ out, int G) {
    int idx = blockIdx.x * 256 + threadIdx.x;
    const int HP = KTOP / 2;
    if (idx >= G * 16 * HP) return;
    int g = idx / (16 * HP);
    int rem = idx - g * 16 * HP;
    int f = rem / HP, t = rem - f * HP;
    const float* p = in + (size_t)g * 16 * KTOP + f * KTOP + 2 * t;
    out[(size_t)g * 16 * HP + f * HP + t] = fmaxf(p[0], p[1]);
}

// ---------------- conv1d(16,32,5) + relu -> z [G,352] (f*11+t) ----------------
__global__ void k_conv2(const float* __restrict__ pool, const float* __restrict__ w,
                        const float* __restrict__ b, float* __restrict__ z, int G) {
    int idx = blockIdx.x * 256 + threadIdx.x;
    if (idx >= G * 32 * 11) return;
    int g = idx / (32 * 11);
    int rem = idx - g * 32 * 11;
    int f = rem / 11, t = rem - f * 11;
    const float* pg = pool + (size_t)g * 16 * 15;
    const float* wf = w + f * 16 * 5;
    float s = b[f];
#pragma unroll
    for (int c = 0; c < 16; c++) {
        const float* pc = pg + c * 15 + t;
        const float* wc = wf + c * 5;
#pragma unroll
        for (int i = 0; i < 5; i++) s += pc[i] * wc[i];
    }
    z[(size_t)g * 352 + f * 11 + t] = fmaxf(s, 0.f);
}

// ---------------- FC1 352->128 + relu ----------------
__global__ void k_fc1(const float* __restrict__ z, const float* __restrict__ w,
                      const float* __restrict__ b, float* __restrict__ h, int G) {
    int idx = blockIdx.x * 256 + threadIdx.x;
    if (idx >= G * 128) return;
    int g = idx >> 7, o = idx & 127;
    const float* zg = z + (size_t)g * 352;
    float s = b[o];
    for (int i = 0; i < 352; i++) s += zg[i] * w[i * 128 + o];
    h[idx] = fmaxf(s, 0.f);
}

// ---------------- FC2 128->10 + log_softmax ----------------
__global__ void k_fc2(const float* __restrict__ h, const float* __restrict__ w,
                      const float* __restrict__ b, float* __restrict__ out, int G) {
    int g = blockIdx.x * 256 + threadIdx.x;
    if (g >= G) return;
    float l[NCLS];
#pragma unroll
    for (int c = 0; c < NCLS; c++) l[c] = b[c];
    const float* hg = h + (size_t)g * 128;
    for (int i = 0; i < 128; i++) {
        float hv = hg[i];
        const float* wr = w + i * NCLS;
#pragma unroll
        for (int c = 0; c < NCLS; c++) l[c] += hv * wr[c];
    }
    float mx = l[0];
#pragma unroll
    for (int c = 1; c < NCLS; c++) mx = fmaxf(mx, l[c]);
    float ssum = 0.f;
#pragma unroll
    for (int c = 0; c < NCLS; c++) ssum += expf(l[c] - mx);
    float lse = mx + logf(ssum);
#pragma unroll
    for (int c = 0; c < NCLS; c++) out[(size_t)g * NCLS + c] = l[c] - lse;
}

// ---------------- host launch ----------------
extern "C" void kernel_launch(void* const* d_in, const int* in_sizes, int n_in,
                              void* d_out, int out_size, void* d_ws, size_t ws_size,
                              hipStream_t stream) {
    const float* x   = (const float*)d_in[0];
    const int*   src = (const int*)d_in[1];
    const int    E   = in_sizes[1] / 2;
    const int*   dst = src + E;
    const int    N   = in_sizes[0] / F_INPUT;
    const int    G   = N / NPG_C;

    const float* W1 = (const float*)d_in[3];  const float* b1 = (const float*)d_in[4];
    const float* W2 = (const float*)d_in[5];  const float* b2 = (const float*)d_in[6];
    const float* W3 = (const float*)d_in[7];  const float* b3 = (const float*)d_in[8];
    const float* W4 = (const float*)d_in[9];  const float* b4 = (const float*)d_in[10];
    const float* c5w = (const float*)d_in[11]; const float* c5b = (const float*)d_in[12];
    const float* c6w = (const float*)d_in[13]; const float* c6b = (const float*)d_in[14];
    const float* f1w = (const float*)d_in[15]; const float* f1b = (const float*)d_in[16];
    const float* f2w = (const float*)d_in[17]; const float* f2b = (const float*)d_in[18];
    float* out = (float*)d_out;

    // workspace layout (floats)
    float* ws = (float*)d_ws;
    size_t o = 0;
    float* dinv = ws + o; o += (size_t)N;
    float* xd   = ws + o; o += (size_t)N * CH;
    float* H    = ws + o; o += (size_t)N * CH;
    float* agg  = ws + o; o += (size_t)N * CH;
    float* xc   = ws + o; o += (size_t)N * NCAT;
    // reuse retired regions for the post-GCN pipeline
    float* h4    = H;                 // N  (before sort-pool)
    float* agg4  = H + N;             // N
    float* topk  = H;                 // G*KTOP*97 <= N*32
    float* conv1 = agg;               // G*480
    float* pool  = agg + (size_t)G * 16 * KTOP;        // G*240
    float* z     = pool + (size_t)G * 16 * (KTOP / 2); // G*352
    float* fc1   = z + (size_t)G * 352;                // G*128

    const int TB = 256;
    dim3 blk(TB);
    int gN   = (N + TB - 1) / TB;
    int gE   = (E + TB - 1) / TB;
    int gNC  = (N * CH + TB - 1) / TB;
    int gGem = (N + 127) / 128;

    // degrees -> dinv (in place)
    k_zero<<<gN, blk, 0, stream>>>(dinv, N);
    k_deg<<<gE, blk, 0, stream>>>(dst, dinv, E);
    k_dinv<<<gN, blk, 0, stream>>>(dinv, N);

    // GCN layer 1 (K=128)
    k_gemm_n32<<<gGem, blk, 0, stream>>>(x, W1, H, N, F_INPUT);
    k_init32<<<gNC, blk, 0, stream>>>(H, dinv, agg, N);
    k_scatter32<<<gE, blk, 0, stream>>>(src, dst, dinv, H, agg, E);
    k_fin32<<<gNC, blk, 0, stream>>>(agg, b1, xd, xc, N, 0);

    // GCN layer 2 (K=32)
    k_gemm_n32<<<gGem, blk, 0, stream>>>(xd, W2, H, N, CH);
    k_init32<<<gNC, blk, 0, stream>>>(H, dinv, agg, N);
    k_scatter32<<<gE, blk, 0, stream>>>(src, dst, dinv, H, agg, E);
    k_fin32<<<gNC, blk, 0, stream>>>(agg, b2, xd, xc, N, 32);

    // GCN layer 3 (K=32)
    k_gemm_n32<<<gGem, blk, 0, stream>>>(xd, W3, H, N, CH);
    k_init32<<<gNC, blk, 0, stream>>>(H, dinv, agg, N);
    k_scatter32<<<gE, blk, 0, stream>>>(src, dst, dinv, H, agg, E);
    k_fin32<<<gNC, blk, 0, stream>>>(agg, b3, xd, xc, N, 64);

    // GCN layer 4 (1 channel)
    k_h4<<<gN, blk, 0, stream>>>(xd, W4, h4, N);
    k_init4<<<gN, blk, 0, stream>>>(h4, dinv, agg4, N);
    k_scatter4<<<gE, blk, 0, stream>>>(src, dst, dinv, h4, agg4, E);
    k_fin4<<<gN, blk, 0, stream>>>(agg4, b4, xc, N);

    // sort-pool -> topk [G, 30, 97]
    k_sortpool<<<G, dim3(128), 0, stream>>>(xc, topk);

    // conv1 + relu, maxpool, conv2 + relu, fc1 + relu, fc2 + log_softmax
    k_conv1<<<(G * 16 * KTOP + TB - 1) / TB, blk, 0, stream>>>(topk, c5w, c5b, conv1, G);
    k_pool<<<(G * 16 * (KTOP / 2) + TB - 1) / TB, blk, 0, stream>>>(conv1, pool, G);
    k_conv2<<<(G * 32 * 11 + TB - 1) / TB, blk, 0, stream>>>(pool, c6w, c6b, z, G);
    k_fc1<<<(G * 128 + TB - 1) / TB, blk, 0, stream>>>(z, f1w, f1b, fc1, G);
    k_fc2<<<(G + TB - 1) / TB, blk, 0, stream>>>(fc1, f2w, f2b, out, G);
}